// PairwiseRankingLoss_9783935500651
// MI455X (gfx1250) — compile-verified
//
#include <hip/hip_runtime.h>

// Wave32 CDNA5 (gfx1250). One wave = one batch row of 128 scores.
// Masked hinge tile computed as relu( WMMA_f32_16x16x4( A(rank-2), B(rank-2) ) ).

typedef __attribute__((ext_vector_type(2))) float v2f;
typedef __attribute__((ext_vector_type(8))) float v8f;

#define MARGIN 1.0f
#define NCARDS 128
#define TILES  8   // 128 / 16

__global__ __launch_bounds__(256) void pair_hinge_wmma_kernel(
    const float* __restrict__ scores,
    const int*   __restrict__ top_mask,
    const int*   __restrict__ valid_mask,
    float*       __restrict__ partial_loss,
    float*       __restrict__ partial_count,
    int batch)
{
    const int lane = threadIdx.x & 31;
    const int wave_in_block = threadIdx.x >> 5;
    const int b = blockIdx.x * 8 + wave_in_block;   // batch row handled by this wave
    if (b >= batch) return;

    const int   col     = lane & 15;
    const float lo_half = (lane < 16) ? 1.0f : 0.0f;  // zero K=2,3 slots (lanes 16-31)

    const float* srow = scores     + (size_t)b * NCARDS;
    const int*   trow = top_mask   + (size_t)b * NCARDS;
    const int*   vrow = valid_mask + (size_t)b * NCARDS;

    // Per 16-wide chunk X: A-columns (anchor side, mask folded in) and
    // B-rows (candidate side, margin + mask folded in). Valid in lanes 0-15.
    float A0[TILES], A1[TILES], B0[TILES], B1[TILES];
    float t_part = 0.0f, nt_part = 0.0f;

#pragma unroll
    for (int X = 0; X < TILES; ++X) {
        float s = srow[X * 16 + col];
        int   t = trow[X * 16 + col];
        int   v = vrow[X * 16 + col];
        float wt  = (t == 1)             ? lo_half : 0.0f;  // anchor (top) mask
        float wnt = ((t == 0) && (v == 1)) ? lo_half : 0.0f; // non-top valid mask
        A0[X] = s * wt;               // K=0 column of A
        A1[X] = wt;                   // K=1 column of A
        B0[X] = -wnt;                 // K=0 row of B
        B1[X] = wnt * (s + MARGIN);   // K=1 row of B
        t_part  += wt;
        nt_part += wnt;
    }

    // acc holds the running masked-hinge sums in D-matrix layout (8 VGPRs).
    v8f acc = {};

#pragma unroll
    for (int I = 0; I < TILES; ++I) {
        v2f a;
        a.x = A0[I];   // A[M=lane][K=0]  (lanes 16-31 are zero -> K=2 dead)
        a.y = A1[I];   // A[M=lane][K=1]  (lanes 16-31 are zero -> K=3 dead)
#pragma unroll
        for (int J = 0; J < TILES; ++J) {
            v2f bb;
            bb.x = B0[J];  // B[K=0][N=lane]
            bb.y = B1[J];  // B[K=1][N=lane]
            v8f c = {};
            // D[m][n] = wt[m]*wnt[n]*(MARGIN - s_i[m] + s_j[n])
            v8f d = __builtin_amdgcn_wmma_f32_16x16x4_f32(
                false, a, false, bb, (short)0, c, false, false);
#pragma unroll
            for (int r = 0; r < 8; ++r)
                acc[r] += fmaxf(d[r], 0.0f);   // relu == masked hinge (masks are 0/1)
        }
    }

    // Lane-local sum of the 8 D rows.
    float lsum = 0.0f;
#pragma unroll
    for (int r = 0; r < 8; ++r) lsum += acc[r];

    // Wave32 cross-lane reduction.
#pragma unroll
    for (int off = 16; off > 0; off >>= 1) {
        lsum    += __shfl_xor(lsum,    off, 32);
        t_part  += __shfl_xor(t_part,  off, 32);
        nt_part += __shfl_xor(nt_part, off, 32);
    }

    if (lane == 0) {
        partial_loss[b]  = lsum;
        partial_count[b] = t_part * nt_part;  // separable pair count
    }
}

__global__ __launch_bounds__(256) void finalize_kernel(
    const float* __restrict__ partial_loss,
    const float* __restrict__ partial_count,
    float*       __restrict__ out,
    int batch)
{
    __shared__ float sl[256];
    __shared__ float sc[256];
    const int tid = threadIdx.x;

    float L = 0.0f, C = 0.0f;
    for (int i = tid; i < batch; i += 256) {
        L += partial_loss[i];
        C += partial_count[i];
    }
    sl[tid] = L;
    sc[tid] = C;
    __syncthreads();

#pragma unroll
    for (int s = 128; s > 0; s >>= 1) {
        if (tid < s) {
            sl[tid] += sl[tid + s];
            sc[tid] += sc[tid + s];
        }
        __syncthreads();
    }

    if (tid == 0) {
        float loss = sl[0];
        float cnt  = sc[0];
        out[0] = (cnt > 0.0f) ? (loss / fmaxf(cnt, 1.0f)) : loss;
    }
}

extern "C" void kernel_launch(void* const* d_in, const int* in_sizes, int n_in,
                              void* d_out, int out_size, void* d_ws, size_t ws_size,
                              hipStream_t stream) {
    const float* scores     = (const float*)d_in[0];
    const int*   top_mask   = (const int*)d_in[1];
    const int*   valid_mask = (const int*)d_in[2];

    const int batch = in_sizes[0] / NCARDS;   // 4096 for the reference shapes

    float* partial_loss  = (float*)d_ws;
    float* partial_count = partial_loss + batch;

    // 8 waves (batches) per 256-thread block.
    const int blocks = (batch + 7) / 8;
    pair_hinge_wmma_kernel<<<blocks, 256, 0, stream>>>(
        scores, top_mask, valid_mask, partial_loss, partial_count, batch);

    finalize_kernel<<<1, 256, 0, stream>>>(
        partial_loss, partial_count, (float*)d_out, batch);
}